// SetCriterion_6743098655401
// MI455X (gfx1250) — compile-verified
//
#include <hip/hip_runtime.h>

// ---------------------------------------------------------------------------
// CDNA5 / gfx1250. wave32.  WMMA f32 16x16x4 used as a 32-lane all-reduce.
// ---------------------------------------------------------------------------

typedef float v2f __attribute__((ext_vector_type(2)));
typedef float v8f __attribute__((ext_vector_type(8)));

#define PI_F 3.14159265358979323846f

// 32-lane all-reduce (sum) using two chained V_WMMA_F32_16X16X4_F32.
// A-matrix 16x4 layout (ISA 7.12.2): lanes 0-15 -> (M=lane, K=0/1 in VGPR0/1),
// lanes 16-31 -> (M=lane-16, K=2/3). With a[0]=p, a[1]=0 and B=ones(4x16):
//   D[m,n] = p_m + p_{m+16}  (independent of n)
// D layout: lane L<16: VGPR r = D[r, L]; lane L>=16: VGPR r = D[8+r, L-16].
// Sum of the 8 D regs => lanes 0-15 hold S_lo = sum_{m<8}(p_m+p_{m+16}),
// lanes 16-31 hold S_hi = sum_{m>=8}(...). Second WMMA with a[0]=that sum
// gives every D element = S_lo + S_hi = total, in every lane.
__device__ __forceinline__ float wave_allreduce_sum(float p) {
  v2f a;    a[0] = p;    a[1] = 0.0f;
  v2f ones; ones[0] = 1.0f; ones[1] = 1.0f;
  v8f c = {};
  v8f d = __builtin_amdgcn_wmma_f32_16x16x4_f32(false, a, false, ones,
                                                (short)0, c, false, false);
  float s = ((d[0] + d[1]) + (d[2] + d[3])) + ((d[4] + d[5]) + (d[6] + d[7]));
  v2f a2; a2[0] = s; a2[1] = 0.0f;
  v8f d2 = __builtin_amdgcn_wmma_f32_16x16x4_f32(false, a2, false, ones,
                                                 (short)0, c, false, false);
  return d2[0];
}

// Build R = Rz(t_neg) * Rx(e) * Rz(a_neg) entries from HALF-angle sin/cos of
// the original euler angles (double-angle identities), matching
// _compute_rot_mats(rot) with a=-rot0, e=pi/2+rot1, t=-rot2:
//   sin(a)=-2 sr cr, cos(a)=2 cr^2-1
//   sin(e)=cos(rot1)=1-2 sp^2, cos(e)=-sin(rot1)=-2 sp cp
//   sin(t)=-2 sy cy, cos(t)=1-2 sy^2
__device__ __forceinline__ void build_rotmat(float sr, float cr, float sp, float cp,
                                             float sy, float cy, float R[9]) {
  float sa = -2.0f * sr * cr;
  float ca = 2.0f * cr * cr - 1.0f;
  float se = 1.0f - 2.0f * sp * sp;
  float ce = -2.0f * sp * cp;
  float st = -2.0f * sy * cy;
  float ct = 1.0f - 2.0f * sy * sy;
  // M1 = Rx*Rz ; R = Rz2*M1
  R[0] = ct * ca - st * ce * sa;
  R[1] = -ct * sa - st * ce * ca;
  R[2] = st * se;
  R[3] = st * ca + ct * ce * sa;
  R[4] = -st * sa + ct * ce * ca;
  R[5] = -ct * se;
  R[6] = se * sa;
  R[7] = se * ca;
  R[8] = ce;
}

__global__ __launch_bounds__(256) void crit_main(
    const float* __restrict__ pose_pred, const float* __restrict__ pose_gt,
    const float* __restrict__ trans_pred, const float* __restrict__ trans_gt,
    const float4* __restrict__ lat_pred, const float4* __restrict__ lat_gt,
    float* __restrict__ rot_dist, float* __restrict__ trans_dist,
    float* __restrict__ blk_partials, int n) {
  const int tid = blockIdx.x * blockDim.x + threadIdx.x;
  const int stride = gridDim.x * blockDim.x;

  float acc_q = 0.0f;  // sum of squared quaternion diffs
  float acc_t = 0.0f;  // sum of squared trans diffs
  float acc_l = 0.0f;  // sum of squared latent diffs

  // ---- per-row pose / trans work (one row per thread per iteration) ----
  for (int i = tid; i < n; i += stride) {
    float hp0 = pose_pred[3 * i + 0] * (PI_F * 0.5f);
    float hp1 = pose_pred[3 * i + 1] * (PI_F * 0.5f);
    float hp2 = pose_pred[3 * i + 2] * (PI_F * 0.5f);
    float hg0 = pose_gt[3 * i + 0] * (PI_F * 0.5f);
    float hg1 = pose_gt[3 * i + 1] * (PI_F * 0.5f);
    float hg2 = pose_gt[3 * i + 2] * (PI_F * 0.5f);

    float srp, crp, spp, cpp_, syp, cyp;
    float srg, crg, spg, cpg, syg, cyg;
    __sincosf(hp0, &srp, &crp);
    __sincosf(hp1, &spp, &cpp_);
    __sincosf(hp2, &syp, &cyp);
    __sincosf(hg0, &srg, &crg);
    __sincosf(hg1, &spg, &cpg);
    __sincosf(hg2, &syg, &cyg);

    // quaternions
    float qxp = srp * cpp_ * cyp - crp * spp * syp;
    float qyp = crp * spp * cyp + srp * cpp_ * syp;
    float qzp = crp * cpp_ * syp - srp * spp * cyp;
    float qwp = crp * cpp_ * cyp + srp * spp * syp;
    float qxg = srg * cpg * cyg - crg * spg * syg;
    float qyg = crg * spg * cyg + srg * cpg * syg;
    float qzg = crg * cpg * syg - srg * spg * cyg;
    float qwg = crg * cpg * cyg + srg * spg * syg;
    float dqx = qxp - qxg, dqy = qyp - qyg, dqz = qzp - qzg, dqw = qwp - qwg;
    acc_q += dqx * dqx + dqy * dqy + dqz * dqz + dqw * dqw;

    // rotation matrices + geodesic distance
    float Rp[9], Rg[9];
    build_rotmat(srp, crp, spp, cpp_, syp, cyp, Rp);
    build_rotmat(srg, crg, spg, cpg, syg, cyg, Rg);
    float tr = Rp[0] * Rg[0] + Rp[1] * Rg[1] + Rp[2] * Rg[2] +
               Rp[3] * Rg[3] + Rp[4] * Rg[4] + Rp[5] * Rg[5] +
               Rp[6] * Rg[6] + Rp[7] * Rg[7] + Rp[8] * Rg[8];
    float cang = fminf(fmaxf((tr - 1.0f) * 0.5f, -1.0f), 1.0f);
    rot_dist[i] = acosf(cang) * (180.0f / PI_F);

    // trans loss + trans distance
    float tp0 = trans_pred[3 * i + 0];
    float tp1 = trans_pred[3 * i + 1];
    float tp2 = trans_pred[3 * i + 2];
    float tg0 = trans_gt[3 * i + 0];
    float tg1 = trans_gt[3 * i + 1];
    float tg2 = trans_gt[3 * i + 2];
    float d0 = tp0 - tg0, d1 = tp1 - tg1, d2 = tp2 - tg2;
    acc_t += d0 * d0 + d1 * d1 + d2 * d2;
    float e1 = tg0 * tg1 - tp0 * tp1;
    float e2 = tg0 * tg2 - tp0 * tp2;
    trans_dist[i] = sqrtf(e1 * e1 + e2 * e2) * 100.0f;
  }

  // ---- latent MSE: flat, fully coalesced float4 stream (dominant traffic) ----
  const int m = n * 16;  // n*64 floats / 4
  for (int j = tid; j < m; j += stride) {
    float4 a = lat_pred[j];
    float4 b = lat_gt[j];
    float dx = a.x - b.x, dy = a.y - b.y, dz = a.z - b.z, dw = a.w - b.w;
    acc_l += dx * dx + dy * dy + dz * dz + dw * dw;
  }

  // ---- deterministic reduction: WMMA wave all-reduce, then LDS across waves ----
  float wq = wave_allreduce_sum(acc_q);
  float wt = wave_allreduce_sum(acc_t);
  float wl = wave_allreduce_sum(acc_l);

  __shared__ float sm[3][8];  // 256 threads = 8 wave32
  int lane = threadIdx.x & 31;
  int wv = threadIdx.x >> 5;
  if (lane == 0) {
    sm[0][wv] = wq;
    sm[1][wv] = wt;
    sm[2][wv] = wl;
  }
  __syncthreads();
  if (threadIdx.x == 0) {
    float a = 0.0f, b = 0.0f, c = 0.0f;
    for (int w = 0; w < 8; ++w) {
      a += sm[0][w];
      b += sm[1][w];
      c += sm[2][w];
    }
    blk_partials[blockIdx.x * 3 + 0] = a;
    blk_partials[blockIdx.x * 3 + 1] = b;
    blk_partials[blockIdx.x * 3 + 2] = c;
  }
}

__global__ __launch_bounds__(32) void crit_final(
    const float* __restrict__ blk_partials, int nblocks, float* __restrict__ out,
    float inv_q, float inv_t, float inv_l) {
  float q = 0.0f, t = 0.0f, l = 0.0f;
  for (int b = threadIdx.x; b < nblocks; b += 32) {
    q += blk_partials[3 * b + 0];
    t += blk_partials[3 * b + 1];
    l += blk_partials[3 * b + 2];
  }
  q = wave_allreduce_sum(q);
  t = wave_allreduce_sum(t);
  l = wave_allreduce_sum(l);
  if (threadIdx.x == 0) {
    out[0] = q * inv_q;
    out[1] = t * inv_t;
    out[2] = l * inv_l;
  }
}

extern "C" void kernel_launch(void* const* d_in, const int* in_sizes, int n_in,
                              void* d_out, int out_size, void* d_ws, size_t ws_size,
                              hipStream_t stream) {
  const float* pose_pred = (const float*)d_in[0];
  const float* pose_gt = (const float*)d_in[1];
  const float* trans_pred = (const float*)d_in[2];
  const float* trans_gt = (const float*)d_in[3];
  const float4* lat_pred = (const float4*)d_in[4];
  const float4* lat_gt = (const float4*)d_in[5];

  const int n = in_sizes[0] / 3;  // rows

  float* out = (float*)d_out;
  float* rot_dist = out + 3;
  float* trans_dist = out + 3 + n;
  float* blk = (float*)d_ws;  // grid*3 floats = 24 KB

  const int BLOCK = 256;
  const int GRID = 2048;  // n/256: exactly 1 row/thread, 16 latent float4-pairs/thread

  crit_main<<<GRID, BLOCK, 0, stream>>>(pose_pred, pose_gt, trans_pred, trans_gt,
                                        lat_pred, lat_gt, rot_dist, trans_dist,
                                        blk, n);
  crit_final<<<1, 32, 0, stream>>>(blk, GRID, out,
                                   1.0f / (4.0f * (float)n),
                                   1.0f / (3.0f * (float)n),
                                   1.0f / (64.0f * (float)n));
}